// VariationalGCNEncoder_31593779429476
// MI455X (gfx1250) — compile-verified
//
#include <hip/hip_runtime.h>

typedef float v2f __attribute__((ext_vector_type(2)));
typedef float v4f __attribute__((ext_vector_type(4)));
typedef float v8f __attribute__((ext_vector_type(8)));

#define KDIM 256  // inner dimension of every GEMM in this model

static inline unsigned cdiv(size_t a, size_t b) { return (unsigned)((a + b - 1) / b); }

// ---------------- degree + per-dst edge counts ----------------

__global__ __launch_bounds__(256) void init_kernel(float* deg, int* cnt, int n) {
  int i = blockIdx.x * blockDim.x + threadIdx.x;
  if (i < n) { deg[i] = 1.0f; cnt[i] = 0; }   // deg starts at 1 (self-loop)
}

__global__ __launch_bounds__(256) void count_kernel(float* deg, int* cnt,
                                                    const int* __restrict__ dst, int e) {
  int i = blockIdx.x * blockDim.x + threadIdx.x;
  if (i < e) {
    int d = dst[i];
    atomicAdd(&deg[d], 1.0f);
    atomicAdd(&cnt[d], 1);
  }
}

__global__ __launch_bounds__(256) void dinv_kernel(float* deg, int n) {
  int i = blockIdx.x * blockDim.x + threadIdx.x;
  if (i < n) {
    float d = deg[i];
    deg[i] = d > 0.0f ? rsqrtf(d) : 0.0f;  // in place: deg -> dinv
  }
}

// ---------------- single-block exclusive prefix scan (LDS ladder) ----------------

__global__ __launch_bounds__(1024) void scan_kernel(const int* __restrict__ cnt,
                                                    int* __restrict__ off,
                                                    int* __restrict__ cursor, int n) {
  __shared__ int sdata[1024];
  __shared__ int carry;
  if (threadIdx.x == 0) carry = 0;
  __syncthreads();
  for (int base = 0; base < n; base += 1024) {
    int i = base + (int)threadIdx.x;
    int v = (i < n) ? cnt[i] : 0;
    sdata[threadIdx.x] = v;
    __syncthreads();
    for (int s = 1; s < 1024; s <<= 1) {
      int t = (threadIdx.x >= (unsigned)s) ? sdata[threadIdx.x - s] : 0;
      __syncthreads();
      sdata[threadIdx.x] += t;
      __syncthreads();
    }
    int excl = sdata[threadIdx.x] - v;
    if (i < n) { int o = carry + excl; off[i] = o; cursor[i] = o; }
    __syncthreads();                        // everyone has read carry
    if (threadIdx.x == 1023) carry += sdata[1023];
    __syncthreads();
  }
  if (threadIdx.x == 0) off[n] = carry;
}

// ---------------- CSR fill: bucket edges by dst, pre-multiplied norm ----------------

__global__ __launch_bounds__(256) void fill_kernel(const int* __restrict__ src,
                                                   const int* __restrict__ dst,
                                                   const float* __restrict__ dinv,
                                                   int* cursor,
                                                   int* __restrict__ csr_src,
                                                   float* __restrict__ csr_w, int e) {
  int i = blockIdx.x * blockDim.x + threadIdx.x;
  if (i < e) {
    int s = src[i], d = dst[i];
    int p = atomicAdd(&cursor[d], 1);
    csr_src[p] = s;
    csr_w[p]   = dinv[s] * dinv[d];
  }
}

// ---------------- fp32 WMMA GEMM: T[n, out_w] = X[n, 256] @ W[256, out_w] ----------------

__global__ __launch_bounds__(256) void gemm_f32_wmma(const float* __restrict__ X,
                                                     const float* __restrict__ W,
                                                     float* __restrict__ T,
                                                     int tilesM, int out_w, int nRows) {
  const int lane   = threadIdx.x & 31;
  const int waveId = threadIdx.x >> 5;
  const int tileM  = blockIdx.x * 8 + waveId;   // uniform per wave
  if (tileM >= tilesM) return;                  // whole-wave exit, EXEC stays all-ones

  const int half = lane >> 4;
  const int m    = lane & 15;
  const int col0 = blockIdx.y * 64;

  int arow = tileM * 16 + m;
  if (arow >= nRows) arow = nRows - 1;          // safe clamp (N multiple of 16 here)
  const float* __restrict__ xrow = X + (size_t)arow * KDIM;

  v8f acc0 = {}, acc1 = {}, acc2 = {}, acc3 = {};

  for (int k = 0; k < KDIM; k += 4) {
    v2f a = *(const v2f*)(xrow + k + 2 * half);

    const float* __restrict__ wb = W + (size_t)(k + 2 * half) * out_w + col0 + m;
    v2f b0, b1, b2, b3;
    b0.x = wb[0];  b0.y = wb[out_w];
    b1.x = wb[16]; b1.y = wb[out_w + 16];
    b2.x = wb[32]; b2.y = wb[out_w + 32];
    b3.x = wb[48]; b3.y = wb[out_w + 48];

    acc0 = __builtin_amdgcn_wmma_f32_16x16x4_f32(false, a, false, b0, (short)0, acc0, false, false);
    acc1 = __builtin_amdgcn_wmma_f32_16x16x4_f32(false, a, false, b1, (short)0, acc1, false, false);
    acc2 = __builtin_amdgcn_wmma_f32_16x16x4_f32(false, a, false, b2, (short)0, acc2, false, false);
    acc3 = __builtin_amdgcn_wmma_f32_16x16x4_f32(false, a, false, b3, (short)0, acc3, false, false);
  }

  const int rowBase = tileM * 16 + half * 8;
#pragma unroll
  for (int r = 0; r < 8; ++r) {
    int gr = rowBase + r;
    if (gr < nRows) {
      float* trow = T + (size_t)gr * out_w + col0 + m;
      trow[0]  = acc0[r];
      trow[16] = acc1[r];
      trow[32] = acc2[r];
      trow[48] = acc3[r];
    }
  }
}

// ---------------- pull aggregation, layer 1 (F=256), fused self-loop + bias + ReLU ----------------
// One wave per node. 32-edge batches: coalesced index load + __shfl broadcast;
// b128 row gathers; per-lane prefetch of upcoming neighbor rows; zero atomics.

__global__ __launch_bounds__(256) void pull1_kernel(const float* __restrict__ T,
                                                    const int* __restrict__ off,
                                                    const int* __restrict__ csr_src,
                                                    const float* __restrict__ csr_w,
                                                    const float* __restrict__ dinv,
                                                    const float* __restrict__ bias,
                                                    float* __restrict__ OUT, int n) {
  int node = (int)((blockIdx.x * (size_t)blockDim.x + threadIdx.x) >> 5);
  int lane = threadIdx.x & 31;
  if (node >= n) return;

  float di = dinv[node];
  float wself = di * di;

  const v4f* __restrict__ trow = (const v4f*)(T + (size_t)node * 256);
  v4f acc0 = trow[lane]      * wself;   // elements [4*lane .. 4*lane+3]
  v4f acc1 = trow[lane + 32] * wself;   // elements [128+4*lane ..]

  int e0 = off[node], e1 = off[node + 1];
  for (int ebase = e0; ebase < e1; ebase += 32) {
    int idx = ebase + lane;
    int   sv = 0;
    float wv = 0.0f;
    if (idx < e1) {
      sv = csr_src[idx];
      wv = csr_w[idx];
      __builtin_prefetch(T + (size_t)sv * 256, 0, 1);  // global_prefetch_b8 on my row
    }
    int nb = e1 - ebase; if (nb > 32) nb = 32;
    for (int j = 0; j < nb; ++j) {
      int   s = __shfl(sv, j);
      float w = __shfl(wv, j);
      const v4f* __restrict__ srow = (const v4f*)(T + (size_t)s * 256);
      acc0 += srow[lane]      * w;
      acc1 += srow[lane + 32] * w;
    }
  }

  const v4f* __restrict__ bv = (const v4f*)bias;
  v4f o0 = acc0 + bv[lane];
  v4f o1 = acc1 + bv[lane + 32];
#pragma unroll
  for (int c = 0; c < 4; ++c) {
    o0[c] = fmaxf(o0[c], 0.0f);
    o1[c] = fmaxf(o1[c], 0.0f);
  }
  v4f* orow = (v4f*)(OUT + (size_t)node * 256);
  orow[lane]      = o0;
  orow[lane + 32] = o1;
}

// ---------------- pull aggregation, layer 2 (both heads, F=128 each), fused bias ----------------

__global__ __launch_bounds__(256) void pull2_kernel(const float* __restrict__ Tmu,
                                                    const float* __restrict__ Tls,
                                                    const int* __restrict__ off,
                                                    const int* __restrict__ csr_src,
                                                    const float* __restrict__ csr_w,
                                                    const float* __restrict__ dinv,
                                                    const float* __restrict__ bmu,
                                                    const float* __restrict__ bls,
                                                    float* __restrict__ outMu,
                                                    float* __restrict__ outLs, int n) {
  int node = (int)((blockIdx.x * (size_t)blockDim.x + threadIdx.x) >> 5);
  int lane = threadIdx.x & 31;
  if (node >= n) return;

  float di = dinv[node];
  float wself = di * di;

  const v4f* __restrict__ mrow = (const v4f*)(Tmu + (size_t)node * 128);
  const v4f* __restrict__ lrow = (const v4f*)(Tls + (size_t)node * 128);
  v4f amu = mrow[lane] * wself;
  v4f als = lrow[lane] * wself;

  int e0 = off[node], e1 = off[node + 1];
  for (int ebase = e0; ebase < e1; ebase += 32) {
    int idx = ebase + lane;
    int   sv = 0;
    float wv = 0.0f;
    if (idx < e1) {
      sv = csr_src[idx];
      wv = csr_w[idx];
      __builtin_prefetch(Tmu + (size_t)sv * 128, 0, 1);
      __builtin_prefetch(Tls + (size_t)sv * 128, 0, 1);
    }
    int nb = e1 - ebase; if (nb > 32) nb = 32;
    for (int j = 0; j < nb; ++j) {
      int   s = __shfl(sv, j);
      float w = __shfl(wv, j);
      amu += ((const v4f*)(Tmu + (size_t)s * 128))[lane] * w;
      als += ((const v4f*)(Tls + (size_t)s * 128))[lane] * w;
    }
  }

  v4f omu = amu + ((const v4f*)bmu)[lane];
  v4f ols = als + ((const v4f*)bls)[lane];
  ((v4f*)(outMu + (size_t)node * 128))[lane] = omu;
  ((v4f*)(outLs + (size_t)node * 128))[lane] = ols;
}

// ---------------- driver ----------------

extern "C" void kernel_launch(void* const* d_in, const int* in_sizes, int n_in,
                              void* d_out, int out_size, void* d_ws, size_t ws_size,
                              hipStream_t stream) {
  const float* x   = (const float*)d_in[0];
  const int*   ei  = (const int*)d_in[1];
  const float* W1  = (const float*)d_in[2];
  const float* b1  = (const float*)d_in[3];
  const float* Wmu = (const float*)d_in[4];
  const float* bmu = (const float*)d_in[5];
  const float* Wls = (const float*)d_in[6];
  const float* bls = (const float*)d_in[7];

  const int N = in_sizes[0] / 256;   // 50000
  const int E = in_sizes[1] / 2;     // 800000
  const int HID = 256, OUTC = 128;
  const int* srcIdx = ei;
  const int* dstIdx = ei + E;

  float* out = (float*)d_out;        // [mu (N*128) | logstd (N*128)]

  // workspace layout (256B-aligned slabs)
  char* ws = (char*)d_ws;
  size_t p = 0;
  auto take = [&](size_t bytes) -> char* {
    char* r = ws + p;
    p += (bytes + 255) & ~(size_t)255;
    return r;
  };
  float* dinv    = (float*)take((size_t)N * 4);
  int*   cnt     = (int*)  take((size_t)N * 4);
  int*   off     = (int*)  take(((size_t)N + 1) * 4);
  int*   cursor  = (int*)  take((size_t)N * 4);
  int*   csr_src = (int*)  take((size_t)E * 4);
  float* csr_w   = (float*)take((size_t)E * 4);
  float* t1      = (float*)take((size_t)N * HID * 4);
  float* h       = (float*)take((size_t)N * HID * 4);
  float* tmu     = (float*)take((size_t)N * OUTC * 4);
  float* tls     = (float*)take((size_t)N * OUTC * 4);

  const int tilesM = (N + 15) / 16;

  // 1) symmetric normalization + CSR-by-dst build
  init_kernel <<<cdiv(N, 256), 256, 0, stream>>>(dinv, cnt, N);
  count_kernel<<<cdiv(E, 256), 256, 0, stream>>>(dinv, cnt, dstIdx, E);
  dinv_kernel <<<cdiv(N, 256), 256, 0, stream>>>(dinv, N);
  scan_kernel <<<1, 1024, 0, stream>>>(cnt, off, cursor, N);
  fill_kernel <<<cdiv(E, 256), 256, 0, stream>>>(srcIdx, dstIdx, dinv, cursor, csr_src, csr_w, E);

  // 2) layer 1: t1 = x @ W1 ; h = relu(pull(t1) + b1)
  {
    dim3 g(cdiv(tilesM, 8), HID / 64);
    gemm_f32_wmma<<<g, 256, 0, stream>>>(x, W1, t1, tilesM, HID, N);
  }
  pull1_kernel<<<cdiv((size_t)N * 32, 256), 256, 0, stream>>>(t1, off, csr_src, csr_w, dinv, b1, h, N);

  // 3) layer 2 heads: mu and logstd (one fused pull for both)
  {
    dim3 g(cdiv(tilesM, 8), OUTC / 64);
    gemm_f32_wmma<<<g, 256, 0, stream>>>(h, Wmu, tmu, tilesM, OUTC, N);
    gemm_f32_wmma<<<g, 256, 0, stream>>>(h, Wls, tls, tilesM, OUTC, N);
  }
  pull2_kernel<<<cdiv((size_t)N * 32, 256), 256, 0, stream>>>(
      tmu, tls, off, csr_src, csr_w, dinv, bmu, bls, out, out + (size_t)N * OUTC, N);
}